// Metalayer_11347303596493
// MI455X (gfx1250) — compile-verified
//
#include <hip/hip_runtime.h>
#include <hip/hip_bf16.h>

typedef float v2f __attribute__((ext_vector_type(2)));
typedef float v8f __attribute__((ext_vector_type(8)));

#define N_WG   512
#define NBR    6
#define RESL   256
#define EYSZ   1536
#define TOTALE ((N_WG + NBR) * RESL)   /* 132608 */
#define MDIM   1024
#define MM     (MDIM * MDIM)
#define KVECF  11.810499f              /* 2*pi/0.532 */
#define N0F    1.45f

// ---------------------------------------------------------------------------
// WMMA fp32 tiled GEMM: D = act(alpha * A@B + bias), row-major everywhere.
// Requires Mr%128==0, Nc%128==0, K%16==0 (true for every call site).
// Macro tile 128x128, 8 waves (wave32), each wave owns 32x64 = 2x4 16x16 tiles.
// ---------------------------------------------------------------------------
__global__ void __launch_bounds__(256)
k_gemm_wmma(const float* __restrict__ A, const float* __restrict__ B,
            const float* __restrict__ bias, float* __restrict__ D,
            int Mr, int Nc, int K, float alpha, int relu)
{
    __shared__ float As[128][20];   // [M][K] tile, padded
    __shared__ float Bs[16][132];   // [K][N] tile, padded

    const int tid   = threadIdx.x;
    const int lane  = tid & 31;
    const int wave  = tid >> 5;
    const int waveM = wave >> 1;            // 0..3
    const int waveN = wave & 1;             // 0..1
    const int rowBase = blockIdx.y * 128;
    const int colBase = blockIdx.x * 128;

    v8f acc[2][4];
    for (int i = 0; i < 2; i++)
        for (int j = 0; j < 4; j++)
            for (int e = 0; e < 8; e++) acc[i][j][e] = 0.0f;

    const int aRow   = tid >> 2;   // 0..63 (two rows per thread: +0, +64)
    const int aChunk = tid & 3;    // k chunk of 4 floats
    const int bCol4  = tid & 31;   // 0..31 (float4 column chunk)
    const int bKrow  = tid >> 5;   // 0..7  (two k-rows: +0, +8)

    for (int k0 = 0; k0 < K; k0 += 16) {
#pragma unroll
        for (int r = 0; r < 2; r++) {
            const int row = aRow + r * 64;
            const float4 va = *(const float4*)(A + (size_t)(rowBase + row) * K + k0 + aChunk * 4);
            *(float4*)&As[row][aChunk * 4] = va;
        }
#pragma unroll
        for (int r = 0; r < 2; r++) {
            const int kr = bKrow + r * 8;
            const float4 vb = *(const float4*)(B + (size_t)(k0 + kr) * Nc + colBase + bCol4 * 4);
            *(float4*)&Bs[kr][bCol4 * 4] = vb;
        }
        __syncthreads();

        if (k0 + 16 < K) {  // hint next tiles into cache (global_prefetch_b8)
            __builtin_prefetch(A + (size_t)(rowBase + aRow) * K + k0 + 16, 0, 1);
            __builtin_prefetch(B + (size_t)(k0 + 16 + bKrow) * Nc + colBase + bCol4 * 4, 0, 1);
        }

#pragma unroll
        for (int kc = 0; kc < 4; kc++) {
            // A frag 16x4: lane L holds A[M=L&15][kb], A[M=L&15][kb+1]; kb=(L>>4)*2
            const int kb = kc * 4 + ((lane >> 4) << 1);
            v2f af[2];
#pragma unroll
            for (int mi = 0; mi < 2; mi++) {
                const int row = waveM * 32 + mi * 16 + (lane & 15);
                af[mi] = *(const v2f*)&As[row][kb];
            }
            // B frag 4x16: lane L holds B[kb][N=L&15], B[kb+1][N=L&15]
            v2f bf[4];
#pragma unroll
            for (int ni = 0; ni < 4; ni++) {
                const int col = waveN * 64 + ni * 16 + (lane & 15);
                bf[ni].x = Bs[kb][col];
                bf[ni].y = Bs[kb + 1][col];
            }
#pragma unroll
            for (int mi = 0; mi < 2; mi++)
#pragma unroll
                for (int ni = 0; ni < 4; ni++)
                    acc[mi][ni] = __builtin_amdgcn_wmma_f32_16x16x4_f32(
                        false, af[mi], false, bf[ni], (short)0, acc[mi][ni], false, false);
        }
        __syncthreads();
    }

    // Epilogue: C/D layout — VGPR g: lanes 0-15 row=g, lanes 16-31 row=g+8; col=lane&15
#pragma unroll
    for (int mi = 0; mi < 2; mi++) {
#pragma unroll
        for (int ni = 0; ni < 4; ni++) {
            const int col = colBase + waveN * 64 + ni * 16 + (lane & 15);
#pragma unroll
            for (int g = 0; g < 8; g++) {
                const int row = rowBase + waveM * 32 + mi * 16 + g + ((lane >> 4) << 3);
                if (row < Mr && col < Nc) {
                    float v = alpha * acc[mi][ni][g];
                    if (bias) v += bias[col];
                    if (relu) v = fmaxf(v, 0.0f);
                    D[(size_t)row * Nc + col] = v;
                }
            }
        }
    }
}

// ---------------------------------------------------------------------------
// Small VALU kernels
// ---------------------------------------------------------------------------
__global__ void k_hs(const float* __restrict__ hp, float* __restrict__ hs) {
    int n = blockIdx.x * blockDim.x + threadIdx.x;
    if (n >= N_WG) return;
    float s = 1.0f / (1.0f + __expf(-hp[n]));
    hs[n] = s * 0.12f + 0.12f;   // H_MIN + s*(H_MAX-H_MIN)
}

// 1 -> 64 -> 64 -> 2 MLP (neff / u0n nets), one sample per thread
__global__ void k_mlp1(const float* __restrict__ hs,
                       const float* __restrict__ W0, const float* __restrict__ b0,
                       const float* __restrict__ W1, const float* __restrict__ b1,
                       const float* __restrict__ W2, const float* __restrict__ b2,
                       float* __restrict__ out) {
    int n = blockIdx.x * blockDim.x + threadIdx.x;
    if (n >= N_WG) return;
    float x = hs[n];
    float h[64];
    for (int j = 0; j < 64; j++) h[j] = fmaxf(x * W0[j] + b0[j], 0.0f);
    float a0 = b2[0], a1 = b2[1];
    for (int j = 0; j < 64; j++) {
        float s = b1[j];
        for (int k = 0; k < 64; k++) s += h[k] * W1[k * 64 + j];
        s = fmaxf(s, 0.0f);
        a0 += s * W2[j * 2 + 0];
        a1 += s * W2[j * 2 + 1];
    }
    out[n * 2 + 0] = a0;
    out[n * 2 + 1] = a1;
}

// Ey net layer 1: 1 -> 128, elementwise
__global__ void k_elayer1(const float* __restrict__ hs, const float* __restrict__ W0,
                          const float* __restrict__ b0, float* __restrict__ h1) {
    int idx = blockIdx.x * blockDim.x + threadIdx.x;
    if (idx >= N_WG * 128) return;
    int n = idx >> 7, j = idx & 127;
    h1[idx] = fmaxf(hs[n] * W0[j] + b0[j], 0.0f);
}

// U0 = 0.002 * eta_u * <Ey[n,m,:], E0[window(n)]>   (purely real)
__global__ void k_u0(const float* __restrict__ Ey, const float* __restrict__ E0,
                     const float* __restrict__ neffu, float* __restrict__ U0) {
    int nm = blockIdx.x * blockDim.x + threadIdx.x;
    if (nm >= MDIM) return;
    int n = nm >> 1, m = nm & 1;
    const float* ey = Ey + (size_t)n * 3072 + m * EYSZ;
    const float* e0 = E0 + n * RESL + (RESL / 2);
    float s = 0.0f;
    for (int f = 0; f < EYSZ; f++) s += ey[f] * e0[f];
    float nu = neffu[nm];
    float eta = nu * N0F / (nu + N0F);
    U0[nm] = 0.002f * eta * s;   // 2*C_EPS*DX
}

__device__ __forceinline__ void net3_64(float i0, float i1, float i2,
                                        const float* W0, const float* b0,
                                        const float* W1, const float* b1,
                                        const float* W2, const float* b2, float* o) {
    float h[64];
    for (int j = 0; j < 64; j++)
        h[j] = fmaxf(i0 * W0[j] + i1 * W0[64 + j] + i2 * W0[128 + j] + b0[j], 0.0f);
    float a0 = b2[0], a1 = b2[1], a2 = b2[2], a3 = b2[3];
    for (int j = 0; j < 64; j++) {
        float s = b1[j];
        for (int k = 0; k < 64; k++) s += h[k] * W1[k * 64 + j];
        s = fmaxf(s, 0.0f);
        a0 += s * W2[j * 4 + 0]; a1 += s * W2[j * 4 + 1];
        a2 += s * W2[j * 4 + 2]; a3 += s * W2[j * 4 + 3];
    }
    o[0] = a0; o[1] = a1; o[2] = a2; o[3] = a3;
}

// per-edge coupling nets (C and K): 3 -> 64 -> 64 -> 4, sum over mi
__global__ void k_edge_mlp(const float* __restrict__ hs,
    const float* cW0, const float* cb0, const float* cW1, const float* cb1,
    const float* cW2, const float* cb2,
    const float* kW0, const float* kb0, const float* kW1, const float* kb1,
    const float* kW2, const float* kb2,
    float* __restrict__ Cv, float* __restrict__ Kv) {
    int i = blockIdx.x * blockDim.x + threadIdx.x;
    if (i >= N_WG * NBR) return;
    int n = i / NBR, o = i % NBR;
    int off = o - 3;
    int c = n + off; c = c < 0 ? 0 : (c > N_WG - 1 ? N_WG - 1 : c);
    float i0 = hs[n], i1 = hs[c], i2 = fabsf((float)off) * 0.3f;
    float oc[4], ok[4];
    net3_64(i0, i1, i2, cW0, cb0, cW1, cb1, cW2, cb2, oc);
    net3_64(i0, i1, i2, kW0, kb0, kW1, kb1, kW2, kb2, ok);
    Cv[i * 2 + 0] = oc[0] + oc[2]; Cv[i * 2 + 1] = oc[1] + oc[3];
    Kv[i * 2 + 0] = ok[0] + ok[2]; Kv[i * 2 + 1] = ok[1] + ok[3];
}

// COO scatter-add into dense C, K (duplicates from edge clipping -> atomics)
__global__ void k_scatter(const float* __restrict__ Cv, const float* __restrict__ Kv,
                          float* __restrict__ C, float* __restrict__ K) {
    int t = blockIdx.x * blockDim.x + threadIdx.x;
    if (t >= N_WG * NBR * 2) return;
    int i = t >> 1, mj = t & 1;
    int n = i / NBR, o = i % NBR, off = o - 3;
    int c = n + off; c = c < 0 ? 0 : (c > N_WG - 1 ? N_WG - 1 : c);
    int r2 = n * 2 + mj, c2 = c * 2 + mj;
    atomicAdd(&C[(size_t)r2 * MDIM + c2], Cv[i * 2 + mj]);
    atomicAdd(&K[(size_t)r2 * MDIM + c2], Kv[i * 2 + mj]);
}

__global__ void k_fill(float* __restrict__ p, float v, int n) {
    int i = blockIdx.x * blockDim.x + threadIdx.x;
    if (i < n) p[i] = v;
}
__global__ void k_identity(float* __restrict__ p) {
    int i = blockIdx.x * blockDim.x + threadIdx.x;
    if (i >= MM) return;
    p[i] = ((i >> 10) == (i & (MDIM - 1))) ? 1.0f : 0.0f;
}
__global__ void k_jacobi(float* __restrict__ X, const float* __restrict__ C) {
    int i = blockIdx.x * blockDim.x + threadIdx.x;
    if (i >= MDIM) return;
    X[(size_t)i * MDIM + i] = 1.0f / C[(size_t)i * MDIM + i];
}
// Newton-Schulz update: X = 2X - X(CX)
__global__ void k_nsx(float* __restrict__ X, const float* __restrict__ T2) {
    int i = blockIdx.x * blockDim.x + threadIdx.x;
    if (i < MM) X[i] = 2.0f * X[i] - T2[i];
}
// T = bdiag*C + K, bdiag = neffs_flat * KVEC
__global__ void k_form(float* __restrict__ T, const float* __restrict__ C,
                       const float* __restrict__ K, const float* __restrict__ neffs) {
    int idx = blockIdx.x * blockDim.x + threadIdx.x;
    if (idx >= MM) return;
    int r = idx >> 10;
    T[idx] = neffs[r] * KVECF * C[idx] + K[idx];
}
__global__ void k_axpy(float* __restrict__ dst, const float* __restrict__ src, float a, int n) {
    int i = blockIdx.x * blockDim.x + threadIdx.x;
    if (i < n) dst[i] += a * src[i];
}
// complex square of commuting (Re,Im): Re=T1-T2, Im=2*T3
__global__ void k_sqcombine(float* __restrict__ Ere, float* __restrict__ Eim,
                            const float* __restrict__ T1, const float* __restrict__ T2,
                            const float* __restrict__ T3) {
    int i = blockIdx.x * blockDim.x + threadIdx.x;
    if (i >= MM) return;
    Ere[i] = T1[i] - T2[i];
    Eim[i] = 2.0f * T3[i];
}
// Uz = (Ere + i Eim) @ U0 with real U0
__global__ void k_matvec(const float* __restrict__ Ere, const float* __restrict__ Eim,
                         const float* __restrict__ U0, float* __restrict__ Uzre,
                         float* __restrict__ Uzim) {
    int i = blockIdx.x * blockDim.x + threadIdx.x;
    if (i >= MDIM) return;
    float sr = 0.0f, si = 0.0f;
    const float* re = Ere + (size_t)i * MDIM;
    const float* im = Eim + (size_t)i * MDIM;
    for (int j = 0; j < MDIM; j++) { sr += re[j] * U0[j]; si += im[j] * U0[j]; }
    Uzre[i] = sr; Uzim[i] = si;
}
// gather-formulated overlap-add: En[t] = sum_{n,m in window} eta*Ey*Uz
__global__ void k_gen_en(const float* __restrict__ neffs, const float* __restrict__ Ey,
                         const float* __restrict__ Uzre, const float* __restrict__ Uzim,
                         float* __restrict__ out) {
    int t = blockIdx.x * blockDim.x + threadIdx.x;
    if (t >= TOTALE) return;
    float re = 0.0f, im = 0.0f;
    int j = t - (RESL / 2);
    if (j >= 0) {
        int lo = j - (EYSZ - 1);
        int nlo = lo > 0 ? (lo + RESL - 1) >> 8 : 0;
        int nhi = j >> 8; if (nhi > N_WG - 1) nhi = N_WG - 1;
        for (int n = nlo; n <= nhi; n++) {
            int f = j - (n << 8);
            for (int m = 0; m < 2; m++) {
                float ne = neffs[n * 2 + m];
                float eta = ne * N0F / (ne + N0F);
                float a = eta * Ey[(size_t)n * 3072 + m * EYSZ + f];
                re += a * Uzre[n * 2 + m];
                im += a * Uzim[n * 2 + m];
            }
        }
    }
    out[2 * t + 0] = re;
    out[2 * t + 1] = im;
}

// ---------------------------------------------------------------------------
extern "C" void kernel_launch(void* const* d_in, const int* in_sizes, int n_in,
                              void* d_out, int out_size, void* d_ws, size_t ws_size,
                              hipStream_t stream) {
    (void)in_sizes; (void)n_in; (void)out_size; (void)ws_size;

    const float* E0       = (const float*)d_in[0];
    const float* h_paras  = (const float*)d_in[1];
    const float* nW0 = (const float*)d_in[2],  *nb0 = (const float*)d_in[3];
    const float* nW1 = (const float*)d_in[4],  *nb1 = (const float*)d_in[5];
    const float* nW2 = (const float*)d_in[6],  *nb2 = (const float*)d_in[7];
    const float* uW0 = (const float*)d_in[8],  *ub0 = (const float*)d_in[9];
    const float* uW1 = (const float*)d_in[10], *ub1 = (const float*)d_in[11];
    const float* uW2 = (const float*)d_in[12], *ub2 = (const float*)d_in[13];
    const float* eW0 = (const float*)d_in[14], *eb0 = (const float*)d_in[15];
    const float* eW1 = (const float*)d_in[16], *eb1 = (const float*)d_in[17];
    const float* eW2 = (const float*)d_in[18], *eb2 = (const float*)d_in[19];
    const float* cW0 = (const float*)d_in[20], *cb0 = (const float*)d_in[21];
    const float* cW1 = (const float*)d_in[22], *cb1 = (const float*)d_in[23];
    const float* cW2 = (const float*)d_in[24], *cb2 = (const float*)d_in[25];
    const float* kW0 = (const float*)d_in[26], *kb0 = (const float*)d_in[27];
    const float* kW1 = (const float*)d_in[28], *kb1 = (const float*)d_in[29];
    const float* kW2 = (const float*)d_in[30], *kb2 = (const float*)d_in[31];
    float* out = (float*)d_out;

    float* ws = (float*)d_ws;
    size_t off = 0;
    auto alloc = [&](size_t nfloats) { float* p = ws + off; off += nfloats; return p; };
    float* hs    = alloc(512);
    float* neffs = alloc(1024);
    float* neffu = alloc(1024);
    float* U0    = alloc(1024);
    float* Uzre  = alloc(1024);
    float* Uzim  = alloc(1024);
    float* Cv    = alloc(6144);
    float* Kv    = alloc(6144);
    float* h1    = alloc((size_t)512 * 128);
    float* h2    = alloc((size_t)512 * 128);
    float* Ey    = alloc((size_t)512 * 3072);
    float* Cm    = alloc(MM);
    float* Km    = alloc(MM);
    float* X     = alloc(MM);
    float* T1    = alloc(MM);
    float* T2    = alloc(MM);
    float* T3    = alloc(MM);
    float* S     = alloc(MM);
    float* Pw    = alloc(MM);
    float* Pt    = alloc(MM);
    float* Ere   = alloc(MM);
    float* Eim   = alloc(MM);

    auto gemm = [&](const float* A, const float* B, const float* bias, float* D,
                    int Mr, int Nc, int K, float alpha, int relu) {
        dim3 g(Nc / 128, Mr / 128);
        k_gemm_wmma<<<g, 256, 0, stream>>>(A, B, bias, D, Mr, Nc, K, alpha, relu);
    };

    // 1) hs, per-waveguide scalar nets
    k_hs<<<8, 64, 0, stream>>>(h_paras, hs);
    k_mlp1<<<8, 64, 0, stream>>>(hs, nW0, nb0, nW1, nb1, nW2, nb2, neffs);
    k_mlp1<<<8, 64, 0, stream>>>(hs, uW0, ub0, uW1, ub1, uW2, ub2, neffu);

    // 2) Ey net (layers 2/3 on the matrix engine)
    k_elayer1<<<(512 * 128) / 256, 256, 0, stream>>>(hs, eW0, eb0, h1);
    gemm(h1, eW1, eb1, h2, 512, 128, 128, 1.0f, 1);
    gemm(h2, eW2, eb2, Ey, 512, 3072, 128, 1.0f, 0);

    // 3) input coupling amplitudes U0 (real)
    k_u0<<<16, 64, 0, stream>>>(Ey, E0, neffu, U0);

    // 4) coupling matrices C, K (edge nets + COO scatter)
    k_edge_mlp<<<48, 64, 0, stream>>>(hs, cW0, cb0, cW1, cb1, cW2, cb2,
                                      kW0, kb0, kW1, kb1, kW2, kb2, Cv, Kv);
    k_fill<<<MM / 256, 256, 0, stream>>>(Cm, 0.0f, MM);
    k_fill<<<MM / 256, 256, 0, stream>>>(Km, 0.0f, MM);
    k_scatter<<<24, 256, 0, stream>>>(Cv, Kv, Cm, Km);

    // 5) C^-1 via Newton-Schulz (diagonally dominant C), all GEMMs
    k_fill<<<MM / 256, 256, 0, stream>>>(X, 0.0f, MM);
    k_jacobi<<<16, 64, 0, stream>>>(X, Cm);
    for (int it = 0; it < 16; it++) {
        gemm(Cm, X, nullptr, T1, MDIM, MDIM, MDIM, 1.0f, 0);   // T1 = C X
        gemm(X, T1, nullptr, T2, MDIM, MDIM, MDIM, 1.0f, 0);   // T2 = X C X
        k_nsx<<<MM / 256, 256, 0, stream>>>(X, T2);            // X = 2X - T2
    }

    // 6) S = (WH/2^10) * C^-1 (bdiag*C + K)
    k_form<<<MM / 256, 256, 0, stream>>>(T1, Cm, Km, neffs);
    gemm(X, T1, nullptr, S, MDIM, MDIM, MDIM, 0.6f / 1024.0f, 0);

    // 7) expm(i*S*2^10): Taylor (powers alternate real/imag) then 10 squarings
    k_identity<<<MM / 256, 256, 0, stream>>>(Ere);
    k_fill<<<MM / 256, 256, 0, stream>>>(Eim, 0.0f, MM);
    k_identity<<<MM / 256, 256, 0, stream>>>(Pw);
    for (int k = 1; k <= 16; k++) {
        gemm(Pw, S, nullptr, Pt, MDIM, MDIM, MDIM, 1.0f / (float)k, 0);
        float* tmp = Pw; Pw = Pt; Pt = tmp;
        switch (k & 3) {  // i^k: 1->+i, 2->-1, 3->-i, 0->+1
            case 1: k_axpy<<<MM / 256, 256, 0, stream>>>(Eim, Pw, +1.0f, MM); break;
            case 2: k_axpy<<<MM / 256, 256, 0, stream>>>(Ere, Pw, -1.0f, MM); break;
            case 3: k_axpy<<<MM / 256, 256, 0, stream>>>(Eim, Pw, -1.0f, MM); break;
            default: k_axpy<<<MM / 256, 256, 0, stream>>>(Ere, Pw, +1.0f, MM); break;
        }
    }
    for (int sq = 0; sq < 10; sq++) {   // Ere,Eim commute (polynomials of S)
        gemm(Ere, Ere, nullptr, T1, MDIM, MDIM, MDIM, 1.0f, 0);
        gemm(Eim, Eim, nullptr, T2, MDIM, MDIM, MDIM, 1.0f, 0);
        gemm(Ere, Eim, nullptr, T3, MDIM, MDIM, MDIM, 1.0f, 0);
        k_sqcombine<<<MM / 256, 256, 0, stream>>>(Ere, Eim, T1, T2, T3);
    }

    // 8) Uz = P @ U0 ; overlap-add output field
    k_matvec<<<16, 64, 0, stream>>>(Ere, Eim, U0, Uzre, Uzim);
    k_gen_en<<<(TOTALE + 255) / 256, 256, 0, stream>>>(neffs, Ey, Uzre, Uzim, out);
}